// Bilinear_2894807958007
// MI455X (gfx1250) — compile-verified
//
#include <hip/hip_runtime.h>
#include <hip/hip_bf16.h>
#include <stdint.h>

typedef __attribute__((ext_vector_type(2))) float v2f;
typedef __attribute__((ext_vector_type(8))) float v8f;

#define DFEAT 128
#define PITCH 132   // 128 + 4 pad: row stride of 4 banks -> conflict-free A reads

// ---------------------------------------------------------------------------
// Phase 1: Y = X_source @ W   ([M,128] x [128,128] -> [M,128]), fp32 WMMA.
// W-stationary: each wave owns a 16-column slice of W held in 64 VGPRs for
// the whole kernel. X tiles (16 x 128) staged through padded LDS. Each block
// = 8 waves covers all 128 output columns of one 16-row tile per iteration.
// ---------------------------------------------------------------------------
__global__ __launch_bounds__(256, 1)
void bilinear_gemm_xw(const float* __restrict__ X,
                      const float* __restrict__ W,
                      float* __restrict__ Y,
                      int mtiles) {
    __shared__ float sx[16 * PITCH];

    const int tid  = threadIdx.x;
    const int wave = tid >> 5;        // 0..7 -> 16-col slice of output
    const int lane = tid & 31;
    const int half = lane >> 4;       // 0|1
    const int l16  = lane & 15;
    const int ncol = wave * 16 + l16; // output column owned by this lane

    // Preload W fragments: wf[2*kb + j] = W[kb*4 + j + 2*half][ncol]
    float wf[64];
#pragma unroll
    for (int kb = 0; kb < 32; ++kb) {
#pragma unroll
        for (int j = 0; j < 2; ++j) {
            wf[kb * 2 + j] = W[(kb * 4 + j + 2 * half) * DFEAT + ncol];
        }
    }

    for (int mt = blockIdx.x; mt < mtiles; mt += gridDim.x) {
        const int m_base = mt * 16;

        __syncthreads();  // protect LDS from previous iteration's readers
        // Cooperative load of the 16x128 X tile (512 float4 chunks / 256 thr)
#pragma unroll
        for (int i = tid; i < 512; i += 256) {
            const int row = i >> 5;
            const int c4  = (i & 31) << 2;
            const float4 v = *(const float4*)(X + (size_t)(m_base + row) * DFEAT + c4);
            float* dst = &sx[row * PITCH + c4];
            dst[0] = v.x; dst[1] = v.y; dst[2] = v.z; dst[3] = v.w;
        }
        __syncthreads();

        v8f acc = {};
#pragma unroll
        for (int kb = 0; kb < 32; ++kb) {
            const int k0 = kb * 4 + 2 * half;
            // A fragment: A[l16][k0 + j]  (8B-aligned ds_load_b64)
            const v2f a = *(const v2f*)(&sx[l16 * PITCH + k0]);
            v2f b;
            b[0] = wf[kb * 2 + 0];
            b[1] = wf[kb * 2 + 1];
            acc = __builtin_amdgcn_wmma_f32_16x16x4_f32(
                false, a, false, b, (short)0, acc, false, false);
        }

        // D layout: VGPR j holds D[j + 8*half][l16]
#pragma unroll
        for (int j = 0; j < 8; ++j) {
            Y[(size_t)(m_base + j + 8 * half) * DFEAT + ncol] = acc[j];
        }
    }
}

// ---------------------------------------------------------------------------
// Phase 2: out[e] = dot(Y[src[e]], XT[tgt[e]]) + b.
// One wave (32 lanes) per edge: each lane takes a float4 slice of the 128-d
// vectors (perfectly coalesced 512B per vector per wave), then a 5-step
// wave32 butterfly reduction.
// ---------------------------------------------------------------------------
__global__ __launch_bounds__(256, 1)
void bilinear_edge_dot(const float* __restrict__ Y,
                       const float* __restrict__ XT,
                       const int* __restrict__ eidx,
                       const float* __restrict__ bptr,
                       float* __restrict__ out,
                       int E) {
    const int lane   = threadIdx.x & 31;
    const int gwave  = (int)((blockIdx.x * blockDim.x + threadIdx.x) >> 5);
    const int nwaves = (int)((gridDim.x * blockDim.x) >> 5);
    const float bias = bptr[0];

    for (int e = gwave; e < E; e += nwaves) {
        const int s = eidx[e];       // edge_label_index[0][e]
        const int t = eidx[E + e];   // edge_label_index[1][e]

        const float4 a = *(const float4*)(Y  + (size_t)s * DFEAT + lane * 4);
        const float4 c = *(const float4*)(XT + (size_t)t * DFEAT + lane * 4);

        float sum = a.x * c.x + a.y * c.y + a.z * c.z + a.w * c.w;
        // wave32 butterfly
        sum += __shfl_xor(sum, 16);
        sum += __shfl_xor(sum, 8);
        sum += __shfl_xor(sum, 4);
        sum += __shfl_xor(sum, 2);
        sum += __shfl_xor(sum, 1);

        if (lane == 0) out[e] = sum + bias;
    }
}

// ---------------------------------------------------------------------------
// Inputs (setup_inputs order):
//   0: x_source [N,128] f32   1: x_target [N,128] f32
//   2: edge_label_index [2,E] int   3: W [128,128] f32   4: b scalar f32
// Output: logits [E] f32.  d_ws holds Y = x_source @ W  (N*128 floats).
// ---------------------------------------------------------------------------
extern "C" void kernel_launch(void* const* d_in, const int* in_sizes, int n_in,
                              void* d_out, int out_size, void* d_ws, size_t ws_size,
                              hipStream_t stream) {
    const float* x_source = (const float*)d_in[0];
    const float* x_target = (const float*)d_in[1];
    const int*   eidx     = (const int*)d_in[2];
    const float* W        = (const float*)d_in[3];
    const float* bptr     = (const float*)d_in[4];
    float*       out      = (float*)d_out;
    float*       Y        = (float*)d_ws;   // N*128*4 bytes = 51.2 MB

    const int N = in_sizes[0] / DFEAT;
    const int E = in_sizes[2] / 2;
    const int mtiles = N / 16;   // N = 100000 -> exactly 6250 tiles

    // Phase 1: GEMM (W-stationary, ~640 persistent blocks saturate the WGPs
    // while keeping redundant W preload traffic to ~40 MB of L2 reads).
    int g1 = mtiles < 640 ? mtiles : 640;
    bilinear_gemm_xw<<<g1, 256, 0, stream>>>(x_source, W, Y, mtiles);

    // Phase 2: edge gather + dot (L2-resident gathers; HBM only for edges+out)
    bilinear_edge_dot<<<2048, 256, 0, stream>>>(Y, x_target, eidx, bptr, out, E);
}